// LocationSensitiveAttention_6433861009547
// MI455X (gfx1250) — compile-verified
//
#include <hip/hip_runtime.h>
#include <hip/hip_bf16.h>

// Problem constants (match reference)
#define TT 2048
#define BB 128
#define HH 128
#define QQH 1024
#define NNF 32
#define KKW 31

typedef __attribute__((ext_vector_type(2))) float v2f;
typedef __attribute__((ext_vector_type(8))) float v8f;

// ---------------------------------------------------------------------------
// Kernel 1: qb[b,h] = sum_k query[b,k]*Wq[h,k] + (bq[h] + ba[h] + score_b[h])
// GEMM M=128 (b), N=128 (h), K=1024 via v_wmma_f32_16x16x4_f32.
// Grid: (8,8) tiles, 1 wave (32 threads) per 16x16 tile.
// ---------------------------------------------------------------------------
__global__ __launch_bounds__(32) void qproj_wmma(
    const float* __restrict__ query, const float* __restrict__ Wq,
    const float* __restrict__ bq, const float* __restrict__ ba,
    const float* __restrict__ score_b, float* __restrict__ qb)
{
    const int lane = threadIdx.x;
    const int l15  = lane & 15;
    const int half = lane >> 4;          // 0: K=0,1  1: K=2,3 (A/B f32 16x16x4 layout)
    const int koff = half * 2;
    const int b0 = blockIdx.x * 16;
    const int h0 = blockIdx.y * 16;

    const float* qrow = query + (b0 + l15) * QQH;   // A row (M = b)
    const float* wrow = Wq    + (h0 + l15) * QQH;   // B col (N = h), K contiguous

    v8f c = {};
    for (int k0 = 0; k0 < QQH; k0 += 4) {
        v2f a  = *(const v2f*)(qrow + k0 + koff);
        v2f bv = *(const v2f*)(wrow + k0 + koff);
        c = __builtin_amdgcn_wmma_f32_16x16x4_f32(
                false, a, false, bv, (short)0, c, false, false);
    }

    const int h = h0 + l15;
    const float bias = bq[h] + ba[h] + score_b[h];
#pragma unroll
    for (int r = 0; r < 8; ++r) {
        // C/D layout: vgpr r -> M = r (lanes 0-15) / r+8 (lanes 16-31), N = l15
        const int brow = b0 + r + half * 8;
        qb[brow * HH + h] = c[r] + bias;
    }
}

// ---------------------------------------------------------------------------
// Kernel 2 (fused score pass, one read of encoded_tokens):
//   per block: fixed b, 16 consecutive t.
//   1) LDS conv patch -> loc_feat[16 t][32 f]  (31-tap conv, VALU, tiny)
//   2) per wave (8 waves = 8 h-tiles): [16x32]x[32x16] via 8x WMMA f32 16x16x4
//   3) tanh(enc + qb + loc) . score_w  -> s[b,t] (masked)
// ---------------------------------------------------------------------------
__global__ __launch_bounds__(256) void score_fused(
    const float* __restrict__ enc, const unsigned char* __restrict__ maskp,
    const float* __restrict__ ca,  const float* __restrict__ conv_w,
    const float* __restrict__ conv_b, const float* __restrict__ Wa,
    const float* __restrict__ qb,  const float* __restrict__ score_w,
    float* __restrict__ s_out)
{
    const int b  = blockIdx.y;
    const int t0 = blockIdx.x * 16;
    const int tid = threadIdx.x;

    __shared__ float ca_patch[48];       // ca[b, t0-15 .. t0+30]
    __shared__ float locf[16][33];       // conv output, padded stride
    __shared__ float part[16][8];        // per-wave partial dot sums

    if (tid < 48) {
        const int idx = t0 - 15 + tid;
        ca_patch[tid] = (tid < 46 && idx >= 0 && idx < TT) ? ca[b * TT + idx] : 0.0f;
    }
    __syncthreads();

    // conv: 512 outputs over 256 threads (2 each), 31 FMAs each
    for (int v = tid; v < 16 * NNF; v += 256) {
        const int tl = v >> 5;
        const int f  = v & (NNF - 1);
        const float* w = conv_w + f * KKW;
        float acc = conv_b[f];
#pragma unroll
        for (int k = 0; k < KKW; ++k)
            acc = fmaf(ca_patch[tl + k], w[k], acc);
        locf[tl][f] = acc;
    }
    __syncthreads();

    const int wave = tid >> 5;
    const int lane = tid & 31;
    const int l15  = lane & 15;
    const int half = lane >> 4;
    const int koff = half * 2;
    const int h0   = wave * 16;

    // loc projection: D[t,h] = sum_f locf[t,f] * Wa[h,f]
    const float* warow = Wa + (h0 + l15) * NNF;   // B operand: N = h, K = f contiguous
    v8f c = {};
#pragma unroll
    for (int kc = 0; kc < 8; ++kc) {
        v2f a;
        a.x = locf[l15][kc * 4 + koff];
        a.y = locf[l15][kc * 4 + koff + 1];
        v2f bv = *(const v2f*)(warow + kc * 4 + koff);
        c = __builtin_amdgcn_wmma_f32_16x16x4_f32(
                false, a, false, bv, (short)0, c, false, false);
    }

    const int h   = h0 + l15;
    const float sw = score_w[h];
    const float qv = qb[b * HH + h];

#pragma unroll
    for (int r = 0; r < 8; ++r) {
        const int tl = r + half * 8;                      // D-matrix M index
        const float e = enc[((size_t)(t0 + tl) * BB + b) * HH + h];
        float val = tanhf(c[r] + e + qv);
        float contrib = val * sw;
        // reduce over the 16 lanes sharing this t (halves stay separate)
        contrib += __shfl_xor(contrib, 1);
        contrib += __shfl_xor(contrib, 2);
        contrib += __shfl_xor(contrib, 4);
        contrib += __shfl_xor(contrib, 8);
        if (l15 == 0) part[tl][wave] = contrib;
    }
    __syncthreads();

    if (tid < 16) {
        float acc = 0.0f;
#pragma unroll
        for (int w = 0; w < 8; ++w) acc += part[tid][w];
        const int tg = t0 + tid;
        const bool m = maskp[b * TT + tg] != 0;
        s_out[b * TT + tg] = m ? -__builtin_inff() : acc;
    }
}

// ---------------------------------------------------------------------------
// Kernel 3: row softmax over T, plus cumulative_alignment update.
// One block per b; 256 threads x 8 elements held in registers (single pass).
// ---------------------------------------------------------------------------
__global__ __launch_bounds__(256) void softmax_k(
    const float* __restrict__ s, const float* __restrict__ ca,
    float* __restrict__ align_out, float* __restrict__ cum_out)
{
    const int b = blockIdx.x;
    const int tid = threadIdx.x;
    __shared__ float red[8];

    float sv[8];
#pragma unroll
    for (int i = 0; i < 8; ++i) sv[i] = s[b * TT + tid + i * 256];

    float m = sv[0];
#pragma unroll
    for (int i = 1; i < 8; ++i) m = fmaxf(m, sv[i]);
#pragma unroll
    for (int off = 1; off < 32; off <<= 1) m = fmaxf(m, __shfl_xor(m, off));
    if ((tid & 31) == 0) red[tid >> 5] = m;
    __syncthreads();
    float mall = red[0];
#pragma unroll
    for (int i = 1; i < 8; ++i) mall = fmaxf(mall, red[i]);
    __syncthreads();

    float ev[8];
    float esum = 0.0f;
#pragma unroll
    for (int i = 0; i < 8; ++i) { ev[i] = expf(sv[i] - mall); esum += ev[i]; }
#pragma unroll
    for (int off = 1; off < 32; off <<= 1) esum += __shfl_xor(esum, off);
    if ((tid & 31) == 0) red[tid >> 5] = esum;
    __syncthreads();
    float tot = 0.0f;
#pragma unroll
    for (int i = 0; i < 8; ++i) tot += red[i];

    const float inv = 1.0f / tot;
#pragma unroll
    for (int i = 0; i < 8; ++i) {
        const int t = tid + i * 256;
        const float a = ev[i] * inv;
        align_out[b * TT + t] = a;
        cum_out[b * TT + t]   = ca[b * TT + t] + a;
    }
}

// ---------------------------------------------------------------------------
// Kernel 4: context partials. chunk c covers 128 t's; second pass over enc
// (L2-resident). part[c][b][h] = sum_t align[b,t]*enc[t,b,h]. Fully coalesced.
// ---------------------------------------------------------------------------
__global__ __launch_bounds__(128) void ctx_partial(
    const float* __restrict__ enc, const float* __restrict__ align,
    float* __restrict__ part)
{
    const int c = blockIdx.x;     // 0..15
    const int b = blockIdx.y;     // 0..127
    const int h = threadIdx.x;    // 0..127
    float acc = 0.0f;
    const int tBase = c * 128;
    for (int i = 0; i < 128; ++i) {
        const int t = tBase + i;
        acc = fmaf(align[b * TT + t], enc[((size_t)t * BB + b) * HH + h], acc);
    }
    part[((size_t)c * BB + b) * HH + h] = acc;
}

// Kernel 5: deterministic sequential sum of the 16 chunk partials.
__global__ __launch_bounds__(256) void ctx_reduce(
    const float* __restrict__ part, float* __restrict__ context)
{
    const int idx = blockIdx.x * 256 + threadIdx.x;   // < B*H = 16384
    float acc = 0.0f;
#pragma unroll
    for (int c = 0; c < 16; ++c) acc += part[c * (BB * HH) + idx];
    context[idx] = acc;
}

// ---------------------------------------------------------------------------
extern "C" void kernel_launch(void* const* d_in, const int* in_sizes, int n_in,
                              void* d_out, int out_size, void* d_ws, size_t ws_size,
                              hipStream_t stream)
{
    const float*         enc    = (const float*)d_in[0];          // [T,B,H]
    const unsigned char* maskp  = (const unsigned char*)d_in[1];  // [B,T] bool
    const float*         query  = (const float*)d_in[2];          // [1,B,QH]
    const float*         ca     = (const float*)d_in[3];          // [B,T]
    const float*         conv_w = (const float*)d_in[4];          // [NF,1,K]
    const float*         conv_b = (const float*)d_in[5];          // [NF]
    const float*         Wq     = (const float*)d_in[6];          // [H,QH]
    const float*         bq     = (const float*)d_in[7];          // [H]
    const float*         Wa     = (const float*)d_in[8];          // [H,NF]
    const float*         ba     = (const float*)d_in[9];          // [H]
    const float*         sw     = (const float*)d_in[10];         // [H]
    const float*         sb     = (const float*)d_in[11];         // [H]

    float* out     = (float*)d_out;
    float* context = out;                       // [B,H]   = 16384
    float* cum     = out + BB * HH;             // [B,T]   = 262144
    float* align   = cum + BB * TT;             // [B,T]   = 262144

    float* ws   = (float*)d_ws;
    float* qb   = ws;                           // [B,H]
    float* sbuf = ws + BB * HH;                 // [B,T]
    float* part = sbuf + BB * TT;               // [16,B,H]

    qproj_wmma<<<dim3(BB / 16, HH / 16), 32, 0, stream>>>(query, Wq, bq, ba, sb, qb);
    score_fused<<<dim3(TT / 16, BB), 256, 0, stream>>>(enc, maskp, ca, conv_w, conv_b,
                                                       Wa, qb, sw, sbuf);
    softmax_k<<<BB, 256, 0, stream>>>(sbuf, ca, align, cum);
    ctx_partial<<<dim3(16, BB), 128, 0, stream>>>(enc, align, part);
    ctx_reduce<<<(BB * HH) / 256, 256, 0, stream>>>(part, context);
}